// GNN_37709812859001
// MI455X (gfx1250) — compile-verified
//
#include <hip/hip_runtime.h>
#include <hip/hip_bf16.h>
#include <stdint.h>

#define N_NODES 100000
#define N_EDGES 1600000
#define D_IN    128
#define D_HID   128
#define D_OUT   64

typedef __attribute__((ext_vector_type(2))) float v2f;
typedef __attribute__((ext_vector_type(8))) float v8f;

// ---------------- utility kernels ----------------

__global__ void zero4_k(float4* __restrict__ p, int n4) {
  int i = blockIdx.x * blockDim.x + threadIdx.x;
  if (i < n4) p[i] = make_float4(0.f, 0.f, 0.f, 0.f);
}

__global__ void degree_k(const int* __restrict__ dst, float* __restrict__ deg, int nE) {
  int i = blockIdx.x * blockDim.x + threadIdx.x;
  if (i < nE) unsafeAtomicAdd(&deg[dst[i]], 1.0f);   // exact: integer-valued f32 adds
}

__global__ void invdeg_k(float* __restrict__ deg, int n) {
  int i = blockIdx.x * blockDim.x + threadIdx.x;
  if (i < n) deg[i] = 1.0f / fmaxf(deg[i], 1.0f);
}

// one wave32 per edge; each lane moves 4 of the 128 features (float4 gather,
// 4x global_atomic_add_f32 scatter; agg[] is L2-resident at 51MB)
__global__ void scatter_k(const float* __restrict__ feat,
                          const int* __restrict__ src,
                          const int* __restrict__ dst,
                          float* __restrict__ agg, int nE) {
  int w    = (blockIdx.x * blockDim.x + threadIdx.x) >> 5;
  int lane = threadIdx.x & 31;
  if (w >= nE) return;
  int s = src[w], d = dst[w];
  const float4 v = *(const float4*)(feat + (size_t)s * D_HID + (lane << 2));
  float* o = agg + (size_t)d * D_HID + (lane << 2);
  unsafeAtomicAdd(o + 0, v.x);
  unsafeAtomicAdd(o + 1, v.y);
  unsafeAtomicAdd(o + 2, v.z);
  unsafeAtomicAdd(o + 3, v.w);
}

// ---------------- exact JAX dropout (threefry2x32, key(42) -> (0,42)) --------

__device__ __forceinline__ unsigned rotl32(unsigned x, int n) {
  return (x << n) | (x >> (32 - n));
}

__device__ __forceinline__ float dropout_train(float v, unsigned idx) {
  // jax.random.bernoulli(key(42), 0.5, (N, D_HID)); counts = iota split in half
  const unsigned n   = (unsigned)N_NODES * (unsigned)D_HID;  // 12,800,000
  const unsigned hn  = n >> 1;
  const unsigned k0  = 0u, k1 = 42u;
  const unsigned ks2 = k0 ^ k1 ^ 0x1BD11BDAu;
  const bool lo = idx < hn;
  unsigned X0 = (lo ? idx : idx - hn) + k0;
  unsigned X1 = (lo ? idx + hn : idx) + k1;
#define TFR(r) { X0 += X1; X1 = rotl32(X1, r); X1 ^= X0; }
  TFR(13) TFR(15) TFR(26) TFR(6)  X0 += k1;  X1 += ks2 + 1u;
  TFR(17) TFR(29) TFR(16) TFR(24) X0 += ks2; X1 += k0 + 2u;
  TFR(13) TFR(15) TFR(26) TFR(6)  X0 += k0;  X1 += k1 + 3u;
  TFR(17) TFR(29) TFR(16) TFR(24) X0 += k1;  X1 += ks2 + 4u;
  TFR(13) TFR(15) TFR(26) TFR(6)  X0 += ks2; X1 += k0 + 5u;
#undef TFR
  unsigned bits = lo ? X0 : X1;
  float u = __uint_as_float((bits >> 9) | 0x3f800000u) - 1.0f;  // [0,1)
  return (u < 0.5f) ? v * 2.0f : 0.0f;                          // /(1-p)
}

// ---------------- layer 1: h = dropout(relu(invd*agg @ W1_l + b1 + x @ W1_r))

__launch_bounds__(256)
__global__ void sage_layer1_k(const float* __restrict__ agg,
                              const float* __restrict__ x,
                              const float* __restrict__ Wl,
                              const float* __restrict__ Wr,
                              const float* __restrict__ bias,
                              const float* __restrict__ invd,
                              float* __restrict__ h) {
  const int LDW = 136;                 // 128+8: half-wave bank groups disjoint
  __shared__ float sWl[32 * 136];
  __shared__ float sWr[32 * 136];

  const int wave = threadIdx.x >> 5;
  const int lane = threadIdx.x & 31;
  const int tile = blockIdx.x * 8 + wave;        // 16-row tile
  const bool active = tile < (N_NODES / 16);
  const int half = lane >> 4;                    // K split at midpoint (ISA A 16x4)
  const int mr   = lane & 15;
  const int row  = tile * 16 + mr;

  float idg = 0.f;
  if (active) idg = invd[row];

  v8f acc[8];
  #pragma unroll
  for (int t = 0; t < 8; ++t)
    #pragma unroll
    for (int r = 0; r < 8; ++r) acc[t][r] = 0.0f;

  for (int kc = 0; kc < 4; ++kc) {               // K chunks of 32
    __syncthreads();
    for (int i = threadIdx.x; i < 32 * 32; i += 256) {
      int r = i >> 5, c = (i & 31) << 2;
      size_t g = (size_t)(kc * 32 + r) * D_HID + c;
      *(float4*)&sWl[r * LDW + c] = *(const float4*)&Wl[g];
      *(float4*)&sWr[r * LDW + c] = *(const float4*)&Wr[g];
    }
    __syncthreads();
    if (active) {
      const float* pa = agg + (size_t)row * D_HID + kc * 32 + 2 * half;
      const float* px = x   + (size_t)row * D_HID + kc * 32 + 2 * half;
      for (int kk = 0; kk < 32; kk += 4) {
        v2f aA, aX;
        aA[0] = pa[kk] * idg; aA[1] = pa[kk + 1] * idg;   // fold mean-scale into A
        aX[0] = px[kk];       aX[1] = px[kk + 1];
        const int kr = kk + 2 * half;
        #pragma unroll
        for (int t = 0; t < 8; ++t) {
          const int n = t * 16 + mr;
          v2f bl, br;
          bl[0] = sWl[kr * LDW + n]; bl[1] = sWl[(kr + 1) * LDW + n];
          br[0] = sWr[kr * LDW + n]; br[1] = sWr[(kr + 1) * LDW + n];
          acc[t] = __builtin_amdgcn_wmma_f32_16x16x4_f32(false, aA, false, bl,
                                                         (short)0, acc[t], false, false);
          acc[t] = __builtin_amdgcn_wmma_f32_16x16x4_f32(false, aX, false, br,
                                                         (short)0, acc[t], false, false);
        }
      }
    }
  }

  if (!active) return;
  const int m0 = tile * 16;
  #pragma unroll
  for (int t = 0; t < 8; ++t) {
    const int n = t * 16 + mr;
    const float bn = bias[n];
    #pragma unroll
    for (int r = 0; r < 8; ++r) {
      const int m = m0 + r + 8 * half;           // ISA C/D layout: VGPR r -> M=r / r+8
      float v = fmaxf(acc[t][r] + bn, 0.0f);
      v = dropout_train(v, (unsigned)(m * D_HID + n));
      h[(size_t)m * D_HID + n] = v;
    }
  }
}

// ---------------- layer 2: out = invd*agg @ W2_l + b2 + h @ W2_r -------------

__launch_bounds__(256)
__global__ void sage_layer2_k(const float* __restrict__ agg,
                              const float* __restrict__ hfeat,
                              const float* __restrict__ Wl,
                              const float* __restrict__ Wr,
                              const float* __restrict__ bias,
                              const float* __restrict__ invd,
                              float* __restrict__ out) {
  const int LDW = 80;                  // 64+16: bank-group-disjoint half-waves
  __shared__ float sWl[32 * 80];
  __shared__ float sWr[32 * 80];

  const int wave = threadIdx.x >> 5;
  const int lane = threadIdx.x & 31;
  const int tile = blockIdx.x * 8 + wave;
  const bool active = tile < (N_NODES / 16);
  const int half = lane >> 4;
  const int mr   = lane & 15;
  const int row  = tile * 16 + mr;

  float idg = 0.f;
  if (active) idg = invd[row];

  v8f acc[4];
  #pragma unroll
  for (int t = 0; t < 4; ++t)
    #pragma unroll
    for (int r = 0; r < 8; ++r) acc[t][r] = 0.0f;

  for (int kc = 0; kc < 4; ++kc) {
    __syncthreads();
    for (int i = threadIdx.x; i < 32 * 16; i += 256) {
      int r = i >> 4, c = (i & 15) << 2;
      size_t g = (size_t)(kc * 32 + r) * D_OUT + c;
      *(float4*)&sWl[r * LDW + c] = *(const float4*)&Wl[g];
      *(float4*)&sWr[r * LDW + c] = *(const float4*)&Wr[g];
    }
    __syncthreads();
    if (active) {
      const float* pa = agg   + (size_t)row * D_HID + kc * 32 + 2 * half;
      const float* px = hfeat + (size_t)row * D_HID + kc * 32 + 2 * half;
      for (int kk = 0; kk < 32; kk += 4) {
        v2f aA, aX;
        aA[0] = pa[kk] * idg; aA[1] = pa[kk + 1] * idg;
        aX[0] = px[kk];       aX[1] = px[kk + 1];
        const int kr = kk + 2 * half;
        #pragma unroll
        for (int t = 0; t < 4; ++t) {
          const int n = t * 16 + mr;
          v2f bl, br;
          bl[0] = sWl[kr * LDW + n]; bl[1] = sWl[(kr + 1) * LDW + n];
          br[0] = sWr[kr * LDW + n]; br[1] = sWr[(kr + 1) * LDW + n];
          acc[t] = __builtin_amdgcn_wmma_f32_16x16x4_f32(false, aA, false, bl,
                                                         (short)0, acc[t], false, false);
          acc[t] = __builtin_amdgcn_wmma_f32_16x16x4_f32(false, aX, false, br,
                                                         (short)0, acc[t], false, false);
        }
      }
    }
  }

  if (!active) return;
  const int m0 = tile * 16;
  #pragma unroll
  for (int t = 0; t < 4; ++t) {
    const int n = t * 16 + mr;
    const float bn = bias[n];
    #pragma unroll
    for (int r = 0; r < 8; ++r) {
      const int m = m0 + r + 8 * half;
      out[(size_t)m * D_OUT + n] = acc[t][r] + bn;
    }
  }
}

// ---------------- log_softmax over 64 cols, one wave per row -----------------

__launch_bounds__(256)
__global__ void logsoftmax_k(float* __restrict__ out) {
  int w    = (blockIdx.x * blockDim.x + threadIdx.x) >> 5;
  int lane = threadIdx.x & 31;
  if (w >= N_NODES) return;
  float* row = out + (size_t)w * D_OUT;
  float2 v = *(float2*)(row + lane * 2);
  float m = fmaxf(v.x, v.y);
  #pragma unroll
  for (int o = 16; o > 0; o >>= 1) m = fmaxf(m, __shfl_xor(m, o, 32));
  float s = expf(v.x - m) + expf(v.y - m);
  #pragma unroll
  for (int o = 16; o > 0; o >>= 1) s += __shfl_xor(s, o, 32);
  float lse = m + logf(s);
  v.x -= lse; v.y -= lse;
  *(float2*)(row + lane * 2) = v;
}

// ---------------- host launch ------------------------------------------------

extern "C" void kernel_launch(void* const* d_in, const int* in_sizes, int n_in,
                              void* d_out, int out_size, void* d_ws, size_t ws_size,
                              hipStream_t stream) {
  const float* x   = (const float*)d_in[0];
  const int*   ei  = (const int*)d_in[1];       // (2, E) int32 (JAX x64 disabled)
  const float* W1l = (const float*)d_in[2];
  const float* b1  = (const float*)d_in[3];
  const float* W1r = (const float*)d_in[4];
  const float* W2l = (const float*)d_in[5];
  const float* b2  = (const float*)d_in[6];
  const float* W2r = (const float*)d_in[7];
  const int* src = ei;
  const int* dst = ei + N_EDGES;
  float* outp = (float*)d_out;

  char* ws = (char*)d_ws;
  float* deg = (float*)ws;                                               // N f32
  float* agg = (float*)(ws + (1 << 20));                                 // N*128
  float* h   = (float*)(ws + (1 << 20) +
                        (size_t)N_NODES * D_HID * sizeof(float));        // N*128

  const int B = 256;
  const int aggN4   = N_NODES * D_HID / 4;
  const int gemmBlk = (N_NODES / 16 + 7) / 8;

  // degree + 1/max(deg,1)
  zero4_k<<<(N_NODES / 4 + B - 1) / B, B, 0, stream>>>((float4*)deg, N_NODES / 4);
  degree_k<<<(N_EDGES + B - 1) / B, B, 0, stream>>>(dst, deg, N_EDGES);
  invdeg_k<<<(N_NODES + B - 1) / B, B, 0, stream>>>(deg, N_NODES);

  // layer 1
  zero4_k<<<(aggN4 + B - 1) / B, B, 0, stream>>>((float4*)agg, aggN4);
  scatter_k<<<(N_EDGES + 7) / 8, B, 0, stream>>>(x, src, dst, agg, N_EDGES);
  sage_layer1_k<<<gemmBlk, B, 0, stream>>>(agg, x, W1l, W1r, b1, deg, h);

  // layer 2
  zero4_k<<<(aggN4 + B - 1) / B, B, 0, stream>>>((float4*)agg, aggN4);
  scatter_k<<<(N_EDGES + 7) / 8, B, 0, stream>>>(h, src, dst, agg, N_EDGES);
  sage_layer2_k<<<gemmBlk, B, 0, stream>>>(agg, h, W2l, W2r, b2, deg, outp);

  logsoftmax_k<<<(N_NODES * 32 + B - 1) / B, B, 0, stream>>>(outp);
}